// SoftSkeletonize_16183436771821
// MI455X (gfx1250) — compile-verified
//
#include <hip/hip_runtime.h>
#include <stdint.h>

// ---------------------------------------------------------------------------
// Soft-skeletonize on (2,1,160,224,224) f32, fully fused per term:
//   term t: e = erode_cross(img_t); skel (+)= relu(img_t - dilate3x3x3(e));
//           img_{t+1} = e.   11 terms (t = 0..10), one kernel launch each.
// Memory-bound. Per term: read img 64MB (async->LDS), write e 64MB,
// skel RMW 128MB => 256MB/term, working set ~193MB (mostly in the 192MB L2).
// img staged in LDS with radius-2 halo via CDNA5 async global->LDS engine;
// erode computed into a second LDS tile; dilate taps LDS only.
// Workspace requirement: 2 * 2*160*224*224 * 4 B = 128.5 MB in d_ws.
// ---------------------------------------------------------------------------

#define D_  160
#define H_  224
#define W_  224
#define B_  2
#define HW_ (H_ * W_)                    // 50176
#define DHW_ ((size_t)D_ * H_ * W_)      // 8,028,160

#define TD 4
#define TH 4
#define TW 32
// e tile: radius-1 halo
#define LZ (TD + 2)
#define LY (TH + 2)
#define LX (TW + 2)
#define LN (LZ * LY * LX)                // 1224 floats = 4.9 KB
// img tile: radius-2 halo
#define IZ (TD + 4)
#define IY (TH + 4)
#define IX (TW + 4)
#define IN_ (IZ * IY * IX)               // 2304 floats = 9.2 KB

static_assert(TW == 32 && TH == 4 && TD == 4, "index shifts assume 4x4x32 tile");
static_assert(W_ % TW == 0 && H_ % TH == 0 && D_ % TD == 0, "even tiling");

// ---- CDNA5 async global->LDS copy (ASYNCcnt-tracked) ----------------------
#if defined(__has_builtin)
#if __has_builtin(__builtin_amdgcn_global_load_async_to_lds_b32)
#define ASYNC_LDS 1
#endif
#if __has_builtin(__builtin_amdgcn_s_wait_asynccnt)
#define HAVE_WAIT_ASYNC_BUILTIN 1
#endif
#endif
#ifndef ASYNC_LDS
#define ASYNC_LDS 0
#endif

typedef __attribute__((address_space(1))) int g_int_t;   // global
typedef __attribute__((address_space(3))) int l_int_t;   // LDS

__device__ __forceinline__ void async_copy_f32(const float* __restrict__ g, float* l) {
#if ASYNC_LDS
  __builtin_amdgcn_global_load_async_to_lds_b32(
      (g_int_t*)(uintptr_t)g,
      (l_int_t*)(uint32_t)(uintptr_t)l,
      /*imm offset*/ 0, /*cpol*/ 0);
#else
  *l = *g;
#endif
}

__device__ __forceinline__ void wait_async0() {
#if ASYNC_LDS
#if defined(HAVE_WAIT_ASYNC_BUILTIN)
  __builtin_amdgcn_s_wait_asynccnt(0);
#else
  asm volatile("s_wait_asynccnt 0x0" ::: "memory");
#endif
#endif
}

// ---- fused per-term kernel ------------------------------------------------
template <bool ACC>
__global__ __launch_bounds__(256) void skel_step_k(const float* __restrict__ img,
                                                   float* __restrict__ eout,
                                                   float* __restrict__ skel) {
  __shared__ float it[IN_];   // img tile, halo 2, +inf padded (erode init)
  __shared__ float et[LN];    // erode tile, halo 1, -inf at out-of-volume (dilate init)

  const int tid = (int)threadIdx.x;
  const int x0 = blockIdx.x * TW;
  const int y0 = blockIdx.y * TH;
  const int zb = blockIdx.z;
  const int b  = zb / (D_ / TD);
  const int z0 = (zb % (D_ / TD)) * TD;
  const float* imv = img + (size_t)b * DHW_;

  // 1) stage img tile (async copies for in-range, ds-store pad elsewhere;
  //    slot sets are disjoint so no DS vs ASYNC ordering hazard)
  for (int i = tid; i < IN_; i += 256) {
    const int lx = i % IX;
    const int ly = (i / IX) % IY;
    const int lz = i / (IX * IY);
    const int gx = x0 + lx - 2;
    const int gy = y0 + ly - 2;
    const int gz = z0 + lz - 2;
    if ((unsigned)gx < (unsigned)W_ && (unsigned)gy < (unsigned)H_ &&
        (unsigned)gz < (unsigned)D_) {
      async_copy_f32(imv + (size_t)gz * HW_ + (size_t)gy * W_ + gx, &it[i]);
    } else {
      it[i] = __builtin_inff();
    }
  }
  wait_async0();     // this wave's async copies landed in LDS
  __syncthreads();   // all waves' pad stores + async copies visible

  // 2) erode (7-point cross min) into e tile; -inf where outside the volume
  for (int i = tid; i < LN; i += 256) {
    const int lx = i % LX;
    const int ly = (i / LX) % LY;
    const int lz = i / (LX * LY);
    const int gx = x0 + lx - 1;
    const int gy = y0 + ly - 1;
    const int gz = z0 + lz - 1;
    if ((unsigned)gx < (unsigned)W_ && (unsigned)gy < (unsigned)H_ &&
        (unsigned)gz < (unsigned)D_) {
      const float* c = &it[((lz + 1) * IY + (ly + 1)) * IX + (lx + 1)];
      float m = c[0];
      m = fminf(m, c[-1]);
      m = fminf(m, c[+1]);
      m = fminf(m, c[-IX]);
      m = fminf(m, c[+IX]);
      m = fminf(m, c[-IX * IY]);
      m = fminf(m, c[+IX * IY]);
      et[i] = m;
    } else {
      et[i] = -__builtin_inff();   // dilate's reduce_window init
    }
  }
  __syncthreads();

  // 3) dilate (3x3x3 box max) + write e + accumulate skel
  float* edst = eout + (size_t)b * DHW_;
  float* skv  = skel + (size_t)b * DHW_;
  for (int o = tid; o < TD * TH * TW; o += 256) {
    const int ox = o & (TW - 1);
    const int oy = (o >> 5) & (TH - 1);
    const int oz = o >> 7;
    float d = -__builtin_inff();
#pragma unroll
    for (int dz = 0; dz < 3; ++dz) {
#pragma unroll
      for (int dy = 0; dy < 3; ++dy) {
        const float* t = &et[((oz + dz) * LY + (oy + dy)) * LX + ox];
        d = fmaxf(d, fmaxf(fmaxf(t[0], t[1]), t[2]));
      }
    }
    const float ec  = et[((oz + 1) * LY + (oy + 1)) * LX + (ox + 1)];
    const float imc = it[((oz + 2) * IY + (oy + 2)) * IX + (ox + 2)];
    const size_t g = (size_t)(z0 + oz) * HW_ + (size_t)(y0 + oy) * W_ + (x0 + ox);
    edst[g] = ec;                               // img_{t+1}
    const float r = fmaxf(imc - d, 0.0f);       // relu(img_t - open(img_t))
    skv[g] = ACC ? (skv[g] + r) : r;
  }
}

// ---------------------------------------------------------------------------
extern "C" void kernel_launch(void* const* d_in, const int* in_sizes, int n_in,
                              void* d_out, int out_size, void* d_ws, size_t ws_size,
                              hipStream_t stream) {
  (void)in_sizes; (void)n_in; (void)out_size; (void)ws_size;
  const float* img = (const float*)d_in[0];
  float* skel = (float*)d_out;
  float* eA = (float*)d_ws;                 // erode ping
  float* eB = eA + (size_t)B_ * DHW_;       // erode pong (needs 128.5 MB total ws)

  dim3 grid(W_ / TW, H_ / TH, B_ * (D_ / TD));  // (7, 56, 80)
  dim3 blk(256);

  // term 0: skel = relu(img - open(img)); eA = erode(img) = img_1
  skel_step_k<false><<<grid, blk, 0, stream>>>(img, eA, skel);

  // terms 1..10: skel += relu(img_t - open(img_t)); img_{t+1} ping-pongs
  const float* cur = eA;
  float* nxt = eB;
  for (int i = 0; i < 10; ++i) {
    skel_step_k<true><<<grid, blk, 0, stream>>>(cur, nxt, skel);
    float* t = (float*)cur; cur = nxt; nxt = t;
  }
}